// ProcessSingleEvent_80496277061970
// MI455X (gfx1250) — compile-verified
//
#include <hip/hip_runtime.h>

// ---------------- problem constants (fixed topology) ----------------
#define N_PER_ST   12
#define N_NODES    420
#define MAX_EDGES  5120
#define N_EDGES    4896        // valid edges: 34 station pairs * 144
#define MAX_SUPER  65536
#define N_SUPER    57024       // candidate super-edges: 396 children * 144
#define N_CHILD    396         // nodes 12..407 (stations 1..33)
#define NBLK_S     223         // ceil(57024/256)
#define NBLK_FILL  4096        // persistent blocks for the 2.7GB zero-fill
#define NVEC4      167772160ull // (2*5120*65536)/4 float4 stores

// ---------------- output element offsets (float32, concatenated) ----
static constexpr size_t X_OFF   = 0;                                   // 5120*5
static constexpr size_t RI_OFF  = 25600;                               // 5120*65536
static constexpr size_t RO_OFF  = RI_OFF + (size_t)MAX_EDGES * MAX_SUPER;
static constexpr size_t SUY_OFF = RO_OFF + (size_t)MAX_EDGES * MAX_SUPER;
static constexpr size_t SID_OFF = SUY_OFF + MAX_SUPER;                 // 65536*2
static constexpr size_t NIX_OFF = SID_OFF + 2 * (size_t)MAX_SUPER;     // 5120
static constexpr size_t EID_OFF = NIX_OFF + MAX_EDGES;                 // 5120*3

// ---------------- workspace u32 offsets -----------------------------
#define W_DX  0
#define W_DY  5120
#define W_DZ  10240
#define W_EF  15360              // per-edge flags: bit0 keep, bit1 is_true
#define W_SF  20480              // 57024 super flags
#define W_BS  (20480 + 57024)    // 223 block sums
#define W_BO  (W_BS + 256)       // 223 block exclusive offsets

typedef __attribute__((ext_vector_type(2))) float v2f;
typedef __attribute__((ext_vector_type(4))) float v4f;
typedef __attribute__((ext_vector_type(8))) float v8f;

// ==================== 1) per-edge features ==========================
__global__ void k_edges(const float* __restrict__ ed, float* __restrict__ out,
                        float* __restrict__ wdx, float* __restrict__ wdy,
                        float* __restrict__ wdz, unsigned* __restrict__ wef) {
  int e = blockIdx.x * 256 + threadIdx.x;
  if (e >= MAX_EDGES) return;
  if (e >= N_EDGES) {  // nonzero() padding entries -> masked defaults
    wdx[e] = 0.f; wdy[e] = 0.f; wdz[e] = 0.f; wef[e] = 0u;
    #pragma unroll
    for (int q = 0; q < 5; ++q) out[X_OFF + (size_t)e * 5 + q] = 0.f;
    out[NIX_OFF + e] = -1.0f;
    #pragma unroll
    for (int q = 0; q < 3; ++q) out[EID_OFF + (size_t)e * 3 + q] = -1.0f;
    return;
  }
  int i  = e / 12;
  int jj = e - i * 12;
  int st = i / 12;
  int j  = (st + 1) * 12 + jj;

  const float* pi = ed + (size_t)i * 7;
  const float* pj = ed + (size_t)j * 7;
  float xp = pi[1], yp = pi[2], zp = pi[3], stp = pi[4], tkp = pi[5], iop = pi[6];
  float xc = pj[1], yc = pj[2], zc = pj[3],             tkc = pj[5], ioc = pj[6];

  float rp  = sqrtf(xp * xp + yp * yp);
  float rc  = sqrtf(xc * xc + yc * yc);
  float php = atan2f(xp, yp);   // reference: arctan2(x, y)
  float phc = atan2f(xc, yc);

  const float PI_F  = 3.14159265358979323846f;
  const float TPI_F = 6.28318530717958647692f;
  float d = phc - php;
  if (d >  PI_F) d -= TPI_F;
  if (d < -PI_F) d += TPI_F;
  float dz = zc - zp;
  float dx = rc - rp;

  float ypn = 2.0f * (php + 3.15f) / 6.3f - 1.0f;
  float ycn = 2.0f * (phc + 3.15f) / 6.3f - 1.0f;
  float zpn = 2.0f * (zp + 2386.0f) / 4772.0f - 1.0f;
  float zcn = 2.0f * (zc + 2386.0f) / 4772.0f - 1.0f;
  float dxn = 2.0f * dx / 312.0f;
  float dyn = 2.0f * d  / 6.3f;
  float dzn = 2.0f * dz / 4772.0f;
  float ez  = (stp + 1.0f) / 35.0f;

  bool istrue = (tkp == tkc);
  bool keep = (dyn > -10.0f) && (dyn < 10.0f) && (dzn > -10.0f) && (dzn < 10.0f);

  wdx[e] = dxn; wdy[e] = dyn; wdz[e] = dzn;
  wef[e] = (keep ? 1u : 0u) | (istrue ? 2u : 0u);

  size_t xb = X_OFF + (size_t)e * 5;
  if (keep) {
    out[xb + 0] = ypn; out[xb + 1] = ycn; out[xb + 2] = zpn; out[xb + 3] = zcn; out[xb + 4] = ez;
    out[NIX_OFF + e] = (float)e;
    out[EID_OFF + (size_t)e * 3 + 0] = iop;
    out[EID_OFF + (size_t)e * 3 + 1] = ioc;
    out[EID_OFF + (size_t)e * 3 + 2] = (float)e;
  } else {
    out[xb + 0] = 0.f; out[xb + 1] = 0.f; out[xb + 2] = 0.f; out[xb + 3] = 0.f; out[xb + 4] = 0.f;
    out[NIX_OFF + e] = -1.0f;
    out[EID_OFF + (size_t)e * 3 + 0] = -1.0f;
    out[EID_OFF + (size_t)e * 3 + 1] = -1.0f;
    out[EID_OFF + (size_t)e * 3 + 2] = -1.0f;
  }
}

// ==================== 2a) zero Ri+Ro (2.7 GB stream) ================
// Async LDS->global store engine: each lane streams its private 16B of
// a zeroed 4KB LDS block to global (512B/instruction/wave on ASYNCcnt;
// hardware stalls issue at counter saturation, so no manual pacing).
__global__ void k_fill_riro(float* __restrict__ out) {
  __shared__ __align__(16) float zbuf[1024];          // 4KB of zeros
  int tid = threadIdx.x;
  v4f z = {0.f, 0.f, 0.f, 0.f};
  reinterpret_cast<v4f*>(zbuf)[tid] = z;              // 256*16B = 4KB
  __syncthreads();                                    // DS writes visible

  unsigned lds_off = (unsigned)(size_t)(reinterpret_cast<v4f*>(zbuf) + tid);
  v4f* dst = reinterpret_cast<v4f*>(out + RI_OFF);
  size_t v = (size_t)blockIdx.x * 256 + tid;
  const size_t stride = (size_t)NBLK_FILL * 256;
  for (; v < NVEC4; v += stride) {                    // 160 iterations
    unsigned long long gaddr = (unsigned long long)(dst + v);
    asm volatile("global_store_async_from_lds_b128 %0, %1, off"
                 :: "v"(gaddr), "v"(lds_off) : "memory");
  }
  asm volatile("s_wait_asynccnt 0x0" ::: "memory");   // drain before end
}

// ==================== 2b) default su_y / super_ids ==================
__global__ void k_fill_super(float* __restrict__ out) {
  int k = blockIdx.x * 256 + threadIdx.x;              // 65536 exact
  out[SUY_OFF + k] = 0.0f;
  out[SID_OFF + 2 * (size_t)k + 0] = -2.0f;
  out[SID_OFF + 2 * (size_t)k + 1] = -2.0f;
}

// ==================== 3) pairwise w^2 via WMMA ======================
// One wave per child node c: M[i][t] = ||d(a_i) - d(b_t)||^2 for the
// 12x12 (in-edge x out-edge) block, via aa - 2ab + bb packed into two
// chained v_wmma_f32_16x16x4_f32 ops.
__global__ void k_super_wmma(const float* __restrict__ wdx, const float* __restrict__ wdy,
                             const float* __restrict__ wdz, const unsigned* __restrict__ wef,
                             unsigned* __restrict__ sflags) {
  int wave = (blockIdx.x * 256 + (int)threadIdx.x) >> 5;   // wave-uniform
  int lane = threadIdx.x & 31;
  if (wave >= N_CHILD) return;                              // whole wave exits: EXEC all-1 for WMMA
  int c   = 12 + wave;
  int stc = c / 12;
  int cm  = c - stc * 12;
  int row  = lane & 15;
  int half = lane >> 4;

  // A (16x4): lanes 0-15 hold K={0,1}, lanes 16-31 K={2,3}; row = lane&15
  v2f a1 = {0.f, 0.f}, a2 = {0.f, 0.f};
  if (row < 12) {
    int am = ((stc - 1) * 12 + row) * 12 + cm;              // in-edge a_row
    float ax = wdx[am], ay = wdy[am], az = wdz[am];
    float na2 = ax * ax + ay * ay + az * az;
    if (half == 0) { a1.x = ax; a1.y = ay; a2.x = 1.0f; a2.y = 0.0f; }
    else           { a1.x = az; a1.y = na2; a2.x = 0.0f; a2.y = 0.0f; }
  }
  // B (4x16): VGPR0 rows K={0,2}, VGPR1 rows K={1,3}; col = lane&15
  v2f b1 = {0.f, 0.f}, b2 = {0.f, 0.f};
  int col = lane & 15;
  if (col < 12) {
    int bn = c * 12 + col;                                  // out-edge b_col
    float bx = wdx[bn], by = wdy[bn], bz = wdz[bn];
    float nb2 = bx * bx + by * by + bz * bz;
    if (half == 0) { b1.x = -2.0f * bx; b1.y = -2.0f * by; b2.x = nb2; b2.y = 0.0f; }
    else           { b1.x = -2.0f * bz; b1.y = 1.0f;       b2.x = 0.0f; b2.y = 0.0f; }
  }

  v8f acc = {};
  acc = __builtin_amdgcn_wmma_f32_16x16x4_f32(false, a1, false, b1, (short)0, acc, false, false);
  acc = __builtin_amdgcn_wmma_f32_16x16x4_f32(false, a2, false, b2, (short)0, acc, false, false);

  // C/D layout: VGPR v holds rows v (lanes 0-15) and v+8 (lanes 16-31), col = lane&15
  if (col < 12) {
    int bn = c * 12 + col;
    unsigned keepB = wef[bn] & 1u;
    #pragma unroll
    for (int v = 0; v < 8; ++v) {
      int m = v + 8 * half;
      if (m < 12) {
        int am = ((stc - 1) * 12 + m) * 12 + cm;
        unsigned keepA = wef[am] & 1u;
        float w2 = acc[v];
        sflags[(size_t)am * 12 + col] = keepA & keepB & (w2 < 0.01f ? 1u : 0u);
      }
    }
  }
}

// ==================== 4a) per-block flag sums =======================
__global__ void k_bsums(const unsigned* __restrict__ sflags, unsigned* __restrict__ bsums) {
  __shared__ unsigned sh[256];
  int s = blockIdx.x * 256 + threadIdx.x;
  unsigned f = (s < N_SUPER) ? sflags[s] : 0u;
  sh[threadIdx.x] = f;
  __syncthreads();
  for (int off = 128; off > 0; off >>= 1) {
    if ((int)threadIdx.x < off) sh[threadIdx.x] += sh[threadIdx.x + off];
    __syncthreads();
  }
  if (threadIdx.x == 0) bsums[blockIdx.x] = sh[0];
}

// ==================== 4b) exclusive scan of block sums ==============
__global__ void k_scan(const unsigned* __restrict__ bsums, unsigned* __restrict__ boffs) {
  __shared__ unsigned sh[256];
  int i = threadIdx.x;
  unsigned v = (i < NBLK_S) ? bsums[i] : 0u;
  sh[i] = v;
  __syncthreads();
  for (int off = 1; off < 256; off <<= 1) {
    unsigned t = (i >= off) ? sh[i - off] : 0u;
    __syncthreads();
    sh[i] += t;
    __syncthreads();
  }
  if (i < NBLK_S) boffs[i] = sh[i] - v;   // exclusive
}

// ==================== 4c) compaction + one-hot scatter ==============
__global__ void k_scatter(const unsigned* __restrict__ sflags, const unsigned* __restrict__ boffs,
                          const unsigned* __restrict__ wef, float* __restrict__ out) {
  __shared__ unsigned sh[256];
  int tid = threadIdx.x;
  int s = blockIdx.x * 256 + tid;
  unsigned f = (s < N_SUPER) ? sflags[s] : 0u;
  sh[tid] = f;
  __syncthreads();
  for (int off = 1; off < 256; off <<= 1) {
    unsigned t = (tid >= off) ? sh[tid - off] : 0u;
    __syncthreads();
    sh[tid] += t;
    __syncthreads();
  }
  if (f) {
    unsigned k = boffs[blockIdx.x] + sh[tid] - f;   // global compacted index
    int a  = s / 12;                                // fi (first edge)
    int t  = s - a * 12;
    int ia = a / 12;                                // parent node of a
    int st = ia / 12;
    int cm = a - ia * 12;
    int c  = (st + 1) * 12 + cm;                    // child node of a
    int b  = c * 12 + t;                            // se (second edge)
    float su_true = (((wef[a] >> 1) & (wef[b] >> 1) & 1u)) ? 1.0f : 0.0f;
    out[SUY_OFF + k] = su_true;
    out[SID_OFF + 2 * (size_t)k + 0] = (float)b;    // su_iop = e_index[se]
    out[SID_OFF + 2 * (size_t)k + 1] = (float)a;    // su_ioc = e_index[fi]
    out[RI_OFF + (size_t)a * MAX_SUPER + k] = 1.0f; // Ri one-hot
    out[RO_OFF + (size_t)b * MAX_SUPER + k] = 1.0f; // Ro one-hot
  }
}

// ==================== host launch ===================================
extern "C" void kernel_launch(void* const* d_in, const int* in_sizes, int n_in,
                              void* d_out, int out_size, void* d_ws, size_t ws_size,
                              hipStream_t stream) {
  (void)in_sizes; (void)n_in; (void)out_size; (void)ws_size;
  const float* ed = (const float*)d_in[0];
  float* out = (float*)d_out;
  unsigned* ws = (unsigned*)d_ws;

  float*    wdx    = (float*)(ws + W_DX);
  float*    wdy    = (float*)(ws + W_DY);
  float*    wdz    = (float*)(ws + W_DZ);
  unsigned* wef    = ws + W_EF;
  unsigned* sflags = ws + W_SF;
  unsigned* bsums  = ws + W_BS;
  unsigned* boffs  = ws + W_BO;

  // 1) edge features + small outputs
  k_edges<<<MAX_EDGES / 256, 256, 0, stream>>>(ed, out, wdx, wdy, wdz, wef);

  // 2) bulk zero of Ri+Ro (dominant cost: ~2.7 GB streaming stores),
  //    async LDS->global engine, grid-stride persistent blocks
  k_fill_riro<<<NBLK_FILL, 256, 0, stream>>>(out);
  k_fill_super<<<MAX_SUPER / 256, 256, 0, stream>>>(out);

  // 3) WMMA pairwise distance tiles -> super flags
  k_super_wmma<<<(N_CHILD + 7) / 8, 256, 0, stream>>>(wdx, wdy, wdz, wef, sflags);

  // 4) scan + compact + scatter one-hots
  k_bsums<<<NBLK_S, 256, 0, stream>>>(sflags, bsums);
  k_scan<<<1, 256, 0, stream>>>(bsums, boffs);
  k_scatter<<<NBLK_S, 256, 0, stream>>>(sflags, boffs, wef, out);
}